// WNO1d_19224273616924
// MI455X (gfx1250) — compile-verified
//
#include <hip/hip_runtime.h>
#include <math.h>
#include <cstdint>

// ---------------------------------------------------------------------------
// WNO-1D forward for MI455X (gfx1250, wave32). GEMM-shaped stages (1x1 channel
// conv, fc1) use V_WMMA_F32_16X16X4_F32; shared weight blocks are staged into
// LDS with the gfx1250 async global->LDS path (ASYNCcnt).
// ---------------------------------------------------------------------------

typedef __attribute__((ext_vector_type(2))) float v2f;
typedef __attribute__((ext_vector_type(8))) float v8f;

#define BATCH 128
#define SIZEN 1024
#define WID   64
#define ROWS  (BATCH * WID)      // 8192
#define MODES 14
#define PYRW  1092               // 517+264+137+74+42+26+18+14

#if defined(__has_builtin)
#  if __has_builtin(__builtin_amdgcn_global_load_async_to_lds_b128)
#    define HAVE_ASYNC_LDS 1
#  endif
#endif
#ifndef HAVE_ASYNC_LDS
#  define HAVE_ASYNC_LDS 0
#endif

// Pointer types matching the builtin's signature: b128 payload = <4 x i32>,
// src in addrspace(1) (printed "__device__"), dst in addrspace(3) (LDS).
typedef int vi4 __attribute__((__vector_size__(16)));
typedef __attribute__((address_space(1))) vi4 as1_vi4;
typedef __attribute__((address_space(3))) vi4 as3_vi4;

// Stage 16 bytes global -> LDS. Async path uses ASYNCcnt; fallback is a plain
// vector copy through VGPRs.
__device__ __forceinline__ void stage16(const float* __restrict__ gsrc,
                                        float* __restrict__ lds_dst) {
#if HAVE_ASYNC_LDS
  __builtin_amdgcn_global_load_async_to_lds_b128(
      (as1_vi4*)(uintptr_t)(const void*)gsrc,
      (as3_vi4*)(uint32_t)(uintptr_t)(void*)lds_dst,
      0, 0);
#else
  ((float4*)lds_dst)[0] = ((const float4*)gsrc)[0];
#endif
}

__device__ __forceinline__ void stage_wait() {
#if HAVE_ASYNC_LDS
#  if defined(__has_builtin) && __has_builtin(__builtin_amdgcn_s_wait_asynccnt)
  __builtin_amdgcn_s_wait_asynccnt(0);
#  else
  asm volatile("s_wait_asynccnt 0x0" ::: "memory");
#  endif
#endif
}

__constant__ float DEC_LO_C[12] = {
  -0.00107730108499558f,  0.004777257511010651f, 0.0005538422009938016f,
  -0.031582039318031156f, 0.02752286553001629f,  0.09750160558707936f,
  -0.12976686756709563f, -0.22626469396516913f,  0.3152503517092432f,
   0.7511339080215775f,   0.4946238903983854f,   0.11154074335008017f };
__constant__ float DEC_HI_C[12] = {
  -0.11154074335008017f,  0.4946238903983854f,  -0.7511339080215775f,
   0.3152503517092432f,   0.22626469396516913f, -0.12976686756709563f,
  -0.09750160558707936f,  0.02752286553001629f,  0.031582039318031156f,
   0.0005538422009938016f,-0.004777257511010651f,-0.00107730108499558f };

// ---------------------------------------------------------------------------
// fc0: h[b][c][s] = x[b][s]*W[0][c] + grid(s)*W[1][c] + bias[c]
__global__ void k_fc0(const float* __restrict__ x, const float* __restrict__ w,
                      const float* __restrict__ bias, float* __restrict__ h) {
  int tid = blockIdx.x * blockDim.x + threadIdx.x;
  if (tid >= ROWS * SIZEN) return;
  int s  = tid & (SIZEN - 1);
  int bc = tid >> 10;
  int b  = bc >> 6;
  int c  = bc & 63;
  float xv = x[b * SIZEN + s];
  float g  = (float)s * (1.0f / 1023.0f);
  h[tid] = xv * w[c] + g * w[64 + c] + bias[c];
}

// ---------------------------------------------------------------------------
// DWT analysis level: stride-2 correlation with reversed filters, symmetric pad
__global__ void k_dwt(const float* __restrict__ in, int in_stride, int n,
                      float* __restrict__ lo, int lo_stride,
                      float* __restrict__ hi, int hi_stride,
                      int m, int pl, int total) {
  int tid = blockIdx.x * blockDim.x + threadIdx.x;
  if (tid >= total) return;
  int o   = tid % m;
  int row = tid / m;
  const float* xr = in + (size_t)row * in_stride;
  float alo = 0.f, ahi = 0.f;
#pragma unroll
  for (int t = 0; t < 12; ++t) {
    int j = 2 * o + t - pl;
    while (j < 0 || j >= n) { if (j < 0) j = -j - 1; else j = 2 * n - 1 - j; }
    float v = xr[j];
    alo += v * DEC_LO_C[11 - t];   // H_FWD is filter-reversed
    ahi += v * DEC_HI_C[11 - t];
  }
  lo[(size_t)row * lo_stride + o] = alo;
  hi[(size_t)row * hi_stride + o] = ahi;
}

// ---------------------------------------------------------------------------
// mode mix: out[b][o][x] = sum_i in[b][i][x] * w[i][o][x]  (x < 14)
__global__ void k_mode_mix(const float* __restrict__ in, int in_stride,
                           const float* __restrict__ w, float* __restrict__ out,
                           int total) {
  int tid = blockIdx.x * blockDim.x + threadIdx.x;
  if (tid >= total) return;
  int x  = tid % MODES;
  int bo = tid / MODES;
  int o  = bo & 63;
  int b  = bo >> 6;
  float acc = 0.f;
#pragma unroll 8
  for (int i = 0; i < WID; ++i)
    acc += in[(size_t)(b * WID + i) * in_stride + x] *
           w[((size_t)i * WID + o) * MODES + x];
  out[(size_t)bo * MODES + x] = acc;
}

// ---------------------------------------------------------------------------
// IDWT step: out[j] = sum_t [j+t-1 even, in range] lo[(j+t-1)/2]*LO[t] + hi*HI[t]
__global__ void k_idwt(const float* __restrict__ lo, int lo_stride,
                       const float* __restrict__ hi, int hi_stride,
                       int n, float* __restrict__ out, int out_stride,
                       int outn, int total) {
  int tid = blockIdx.x * blockDim.x + threadIdx.x;
  if (tid >= total) return;
  int j   = tid % outn;
  int row = tid / outn;
  const float* lr = lo + (size_t)row * lo_stride;
  const float* hr = hi + (size_t)row * hi_stride;
  float acc = 0.f;
#pragma unroll
  for (int t = 0; t < 12; ++t) {
    int d = j + t - 1;                       // lhs_dilation=2, pad=(1,1)
    if (d >= 0 && d <= 2 * n - 2 && ((d & 1) == 0)) {
      int k = d >> 1;
      acc += lr[k] * DEC_LO_C[t] + hr[k] * DEC_HI_C[t];
    }
  }
  out[(size_t)row * out_stride + j] = acc;
}

// ---------------------------------------------------------------------------
// 1x1 channel conv via f32 WMMA, fused with bias + wave add + Mish.
// conv_w (16 KB) staged once per block into LDS via async copy. Each wave
// computes a 16(o) x 64(s) strip: one A fragment feeds 4 WMMAs per k-step.
// Tiles: b(128) * ot(4) * sg(16) = 8192 waves, 4 waves/block.
__global__ void k_conv_combine(const float* __restrict__ h,
                               const float* __restrict__ W,
                               const float* __restrict__ bias,
                               const float* __restrict__ wave,
                               float* __restrict__ hout, int act) {
  __shared__ __align__(16) float lW[WID * WID];   // 16 KB
#pragma unroll
  for (int i = 0; i < 8; ++i) {
    int idx4 = i * 128 + threadIdx.x;             // float4 index
    stage16(W + idx4 * 4, lW + idx4 * 4);
  }
  stage_wait();
  __syncthreads();

  int wv   = threadIdx.x >> 5;
  int lane = threadIdx.x & 31;
  int tile = blockIdx.x * 4 + wv;
  int sg = tile & 15;
  int bt = tile >> 4;
  int ot = bt & 3;
  int b  = bt >> 2;
  int half = lane >> 4;
  int l    = lane & 15;
  int o0 = ot * 16;
  int s0 = sg * 64;
  const float* hb = h + (size_t)b * WID * SIZEN + s0 + l;

  v8f cacc[4] = {{}, {}, {}, {}};
#pragma unroll
  for (int k0 = 0; k0 < WID; k0 += 4) {
    int ka = k0 + 2 * half;
    v2f a;                                        // A: M=l, K split by half
    a.x = lW[(o0 + l) * WID + ka];
    a.y = lW[(o0 + l) * WID + ka + 1];
    const float* p0 = hb + (size_t)ka * SIZEN;
    const float* p1 = p0 + SIZEN;
    v2f bm[4];
#pragma unroll
    for (int t = 0; t < 4; ++t) { bm[t].x = p0[t * 16]; bm[t].y = p1[t * 16]; }
#pragma unroll
    for (int t = 0; t < 4; ++t)
      cacc[t] = __builtin_amdgcn_wmma_f32_16x16x4_f32(false, a, false, bm[t],
                                                      (short)0, cacc[t],
                                                      false, false);
  }
#pragma unroll
  for (int t = 0; t < 4; ++t) {
#pragma unroll
    for (int r = 0; r < 8; ++r) {
      int o = o0 + r + 8 * half;                  // D: vgpr r -> M=r / M=r+8
      size_t idx = ((size_t)(b * WID + o)) * SIZEN + s0 + t * 16 + l;
      float v = cacc[t][r] + bias[o] + wave[idx];
      if (act) {                                  // Mish: v * tanh(softplus(v))
        float sp = (v > 20.f) ? v : log1pf(expf(v));
        v = v * tanhf(sp);
      }
      hout[idx] = v;
    }
  }
}

// ---------------------------------------------------------------------------
// fc1 (64->128) + exact GELU + fc2 (128->1), fused. fc1_w (32 KB) staged into
// LDS; per-wave position fragments preloaded into registers once and reused
// across all 8 hidden blocks (128 WMMAs / wave).
__global__ void k_fc12(const float* __restrict__ h, const float* __restrict__ w1,
                       const float* __restrict__ b1, const float* __restrict__ w2,
                       const float* __restrict__ b2, float* __restrict__ out) {
  __shared__ __align__(16) float lW[WID * 128];   // 32 KB
#pragma unroll
  for (int i = 0; i < 16; ++i) {
    int idx4 = i * 128 + threadIdx.x;
    stage16(w1 + idx4 * 4, lW + idx4 * 4);
  }
  stage_wait();
  __syncthreads();

  int wv   = threadIdx.x >> 5;
  int lane = threadIdx.x & 31;
  int tile = blockIdx.x * 4 + wv;                 // 8192 tiles of 16 positions
  int half = lane >> 4;
  int l    = lane & 15;
  int p0 = tile * 16;
  int b  = p0 >> 10;
  int s0 = p0 & (SIZEN - 1);

  // Preload all 16 A fragments (h for these 16 positions, 64 channels).
  v2f afrag[16];
  const float* hbase = h + (size_t)b * WID * SIZEN + s0 + l;
#pragma unroll
  for (int k = 0; k < 16; ++k) {
    int ka = 4 * k + 2 * half;
    afrag[k].x = hbase[(size_t)ka * SIZEN];
    afrag[k].y = hbase[(size_t)(ka + 1) * SIZEN];
  }

  float accum[8];
#pragma unroll
  for (int r = 0; r < 8; ++r) accum[r] = 0.f;

  for (int hb = 0; hb < 8; ++hb) {
    v8f c = {};
    int col = hb * 16 + l;
#pragma unroll
    for (int k = 0; k < 16; ++k) {
      int ka = 4 * k + 2 * half;
      v2f bm;
      bm.x = lW[ka * 128 + col];
      bm.y = lW[(ka + 1) * 128 + col];
      c = __builtin_amdgcn_wmma_f32_16x16x4_f32(false, afrag[k], false, bm,
                                                (short)0, c, false, false);
    }
    float bv  = b1[col];
    float wvv = w2[col];
#pragma unroll
    for (int r = 0; r < 8; ++r) {
      float v = c[r] + bv;
      v = 0.5f * v * (1.0f + erff(v * 0.70710678118654752f));   // exact GELU
      accum[r] += v * wvv;
    }
  }
#pragma unroll
  for (int r = 0; r < 8; ++r) {                   // reduce over N lanes / half
    float v = accum[r];
    v += __shfl_xor(v, 1, 16);
    v += __shfl_xor(v, 2, 16);
    v += __shfl_xor(v, 4, 16);
    v += __shfl_xor(v, 8, 16);
    if (l == 0) out[p0 + r + 8 * half] = v + b2[0];
  }
}

// ---------------------------------------------------------------------------
static inline int cdiv_ll(long long a, long long b) { return (int)((a + b - 1) / b); }

extern "C" void kernel_launch(void* const* d_in, const int* in_sizes, int n_in,
                              void* d_out, int out_size, void* d_ws, size_t ws_size,
                              hipStream_t stream) {
  (void)in_sizes; (void)n_in; (void)out_size; (void)ws_size;
  const float* x       = (const float*)d_in[0];
  const float* fc0_w   = (const float*)d_in[1];
  const float* fc0_b   = (const float*)d_in[2];
  const float* wave_w1 = (const float*)d_in[3];
  const float* wave_w2 = (const float*)d_in[4];
  const float* conv_w  = (const float*)d_in[5];
  const float* conv_b  = (const float*)d_in[6];
  const float* fc1_w   = (const float*)d_in[7];
  const float* fc1_b   = (const float*)d_in[8];
  const float* fc2_w   = (const float*)d_in[9];
  const float* fc2_b   = (const float*)d_in[10];
  float* out = (float*)d_out;

  const size_t R = ROWS;
  float* ws  = (float*)d_ws;
  float* h0  = ws;                 // R*1024
  float* h1  = h0 + R * 1024;      // R*1024
  float* pyr = h1 + R * 1024;      // R*1092  (hi bands, packed per row)
  float* loA = pyr + R * PYRW;     // R*517
  float* loB = loA + R * 517;      // R*264
  float* loM = loB + R * 264;      // R*14
  float* hiM = loM + R * MODES;    // R*14
  float* sA  = hiM + R * MODES;    // R*518  (synthesis ping, even steps)
  float* sB  = sA + R * 518;       // R*1024 (synthesis pong, odd steps)

  { // fc0
    long long total = (long long)R * SIZEN;
    k_fc0<<<cdiv_ll(total, 256), 256, 0, stream>>>(x, fc0_w, fc0_b, h0);
  }

  const int ns[9] = {1024, 517, 264, 137, 74, 42, 26, 18, 14};
  int hoffv[8]; { int a = 0; for (int lv = 0; lv < 8; ++lv) { hoffv[lv] = a; a += ns[lv + 1]; } }

  float* hcur = h0; float* hnext = h1;
  for (int layer = 0; layer < 4; ++layer) {
    const float* w1 = wave_w1 + (size_t)layer * WID * WID * MODES;
    const float* w2 = wave_w2 + (size_t)layer * WID * WID * MODES;
    const float* cw = conv_w + (size_t)layer * WID * WID;
    const float* cb = conv_b + (size_t)layer * WID;

    // ---- analysis: 8 levels, hi bands into pyramid ----
    const float* inb = hcur; int in_stride = SIZEN;
    float* lop[2] = {loA, loB};
    for (int lvl = 0; lvl < 8; ++lvl) {
      int n = ns[lvl], m = ns[lvl + 1];
      int p = 2 * (m - 1) - n + 12, pl = p / 2;
      float* lo = lop[lvl & 1];
      long long total = (long long)R * m;
      k_dwt<<<cdiv_ll(total, 256), 256, 0, stream>>>(
          inb, in_stride, n, lo, m, pyr + hoffv[lvl], PYRW, m, pl, (int)total);
      inb = lo; in_stride = m;
    }

    // ---- mode mixes on 14-length coarsest bands ----
    {
      long long total = (long long)R * MODES;
      k_mode_mix<<<cdiv_ll(total, 256), 256, 0, stream>>>(loB, MODES, w1, loM, (int)total);
      k_mode_mix<<<cdiv_ll(total, 256), 256, 0, stream>>>(pyr + hoffv[7], PYRW, w2, hiM, (int)total);
    }

    // ---- synthesis: 8 levels ----
    const float* lo = loM; int lo_stride = MODES;
    for (int s = 0; s < 8; ++s) {
      int nh = ns[8 - s];            // hi length = samples used (lo truncated)
      int outn = 2 * nh - 10;
      const float* hi; int hi_stride;
      if (s == 0) { hi = hiM; hi_stride = MODES; }
      else        { hi = pyr + hoffv[7 - s]; hi_stride = PYRW; }
      float* outb = (s & 1) ? sB : sA;
      long long total = (long long)R * outn;
      k_idwt<<<cdiv_ll(total, 256), 256, 0, stream>>>(
          lo, lo_stride, hi, hi_stride, nh, outb, outn, outn, (int)total);
      lo = outb; lo_stride = outn;
    }

    // ---- 1x1 conv (WMMA, LDS-staged weights) + bias + wave add + Mish ----
    k_conv_combine<<<2048, 128, 0, stream>>>(hcur, cw, cb, sB, hnext,
                                             (layer < 3) ? 1 : 0);
    float* t = hcur; hcur = hnext; hnext = t;
  }

  // ---- fc1 + GELU + fc2 (WMMA, LDS-staged weights) ----
  k_fc12<<<2048, 128, 0, stream>>>(hcur, fc1_w, fc1_b, fc2_w, fc2_b, out);
}